// STARDNN_43250320671143
// MI455X (gfx1250) — compile-verified
//
#include <hip/hip_runtime.h>
#include <stdint.h>

// ---------------- problem constants ----------------
#define NS        7
#define BATCH     16384
#define MAX_TILES 1031            // 16384/16 + 7 (per-scene pad to mult of 16)
#define MAXP      (MAX_TILES*16)  // 16496 padded rows

typedef __attribute__((ext_vector_type(2))) float v2f;
typedef __attribute__((ext_vector_type(8))) float v8f;
typedef __attribute__((ext_vector_type(4))) int   v4i;

#define AS1 __attribute__((address_space(1)))
#define AS3 __attribute__((address_space(3)))

// ---------------- workspace layout ----------------
#define M_COUNTS 0
#define M_FILL   8
#define M_BASE   16
#define M_NTILES 24
#define M_TSCENE 32
#define M_ROWMAP (M_TSCENE + 1056)               // 1088
#define META_INTS (M_ROWMAP + MAXP)              // 17584 ints = 70336 B
#define H1_OFF    ((size_t)70400)                // 256B aligned
#define H1_BYTES  ((size_t)MAXP * 512 * 4)
#define H2_OFF    (H1_OFF + H1_BYTES)
#define H2_BYTES  ((size_t)MAXP * 256 * 4)
#define WS0_OFF   (H2_OFF + H2_BYTES)            // pre-scaled weights W*gW
#define WS0_BYTES ((size_t)NS * 512 * 512 * 4)
#define WS1_OFF   (WS0_OFF + WS0_BYTES)
#define WS1_BYTES ((size_t)NS * 512 * 256 * 4)
#define WS2_OFF   (WS1_OFF + WS1_BYTES)
#define WS2_BYTES ((size_t)NS * 256 * 64 * 4)

// ---------------- async global->LDS helpers (CDNA5) ----------------
__device__ __forceinline__ void async_b128(const float* gp, float* lp) {
#if __has_builtin(__builtin_amdgcn_global_load_async_to_lds_b128)
  AS1 void* g = (AS1 void*)const_cast<float*>(gp);
  AS3 void* l = (AS3 void*)lp;
  __builtin_amdgcn_global_load_async_to_lds_b128((AS1 v4i*)g, (AS3 v4i*)l, 0, 0);
#else
  uint32_t loff = (uint32_t)(uintptr_t)lp;       // LDS aperture: low 32 = offset
  asm volatile("global_load_async_to_lds_b128 %0, %1, off"
               :: "v"(loff), "v"(gp) : "memory");
#endif
}

__device__ __forceinline__ void wait_async0() {
#if __has_builtin(__builtin_amdgcn_s_wait_asynccnt)
  __builtin_amdgcn_s_wait_asynccnt(0);
#else
  asm volatile("s_wait_asynccnt 0x0" ::: "memory");
#endif
}

// ---------------- bookkeeping kernels ----------------
__global__ void k_init(int* meta) {
  int t = blockIdx.x * blockDim.x + threadIdx.x;
  if (t < 32)   meta[t] = 0;
  if (t < MAXP) meta[M_ROWMAP + t] = -1;
}

__global__ void k_hist(int* meta, const int* __restrict__ scene) {
  int t = blockIdx.x * blockDim.x + threadIdx.x;
  if (t < BATCH) atomicAdd(&meta[M_COUNTS + (scene[t] - 1)], 1);
}

__global__ void k_plan(int* meta) {
  if (threadIdx.x != 0 || blockIdx.x != 0) return;
  int tile = 0;
  for (int s = 0; s < NS; ++s) {
    meta[M_BASE + s] = tile * 16;
    int nt = (meta[M_COUNTS + s] + 15) >> 4;
    for (int i = 0; i < nt; ++i) meta[M_TSCENE + tile + i] = s;
    tile += nt;
  }
  meta[M_NTILES] = tile;
  for (int t2 = tile; t2 < MAX_TILES; ++t2) meta[M_TSCENE + t2] = -1;
}

__global__ void k_scatter(int* meta, const int* __restrict__ scene) {
  int t = blockIdx.x * blockDim.x + threadIdx.x;
  if (t >= BATCH) return;
  int s = scene[t] - 1;
  int slot = atomicAdd(&meta[M_FILL + s], 1);
  meta[M_ROWMAP + meta[M_BASE + s] + slot] = t;
}

// pre-scale weights: Ws[s,k,n] = W[s,k,n] * gW[k,n]
__global__ void k_scale(const float* __restrict__ W, const float* __restrict__ gW,
                        float* __restrict__ Ws, int KN, int total) {
  int t = blockIdx.x * blockDim.x + threadIdx.x;
  if (t < total) Ws[t] = W[t] * gW[t % KN];
}

// ---------------- scene-homogeneous tiled GEMM ----------------
// 128 threads = 4 wave32. Block tile: 16 rows x (64<<TPW) cols.
// Each wave computes 1 or 2 16x16 tiles (shared A fragment) with
// v_wmma_f32_16x16x4_f32. A/B staged in double-buffered LDS via
// global_load_async_to_lds_b128, synced with s_wait_asynccnt.
// All structural flags are template params so the inner loop is branch-free.
template <int TPW, int GATHER, int SCATTER, int RELU>
__global__ __launch_bounds__(128)
void k_star_gemm(const float* __restrict__ A,    // activations
                 const float* __restrict__ Ws,   // pre-scaled (NS, K, N)
                 const float* __restrict__ b,    // (NS, N)
                 const float* __restrict__ gb,   // (N)
                 float* __restrict__ Out,
                 const int* __restrict__ meta,
                 int Kdim, int Ndim) {
  __shared__ float sA[2][16][68];                // pads: 16B-aligned rows
  __shared__ float sB[2][64][132];

  const int tile  = blockIdx.x;
  const int scene = meta[M_TSCENE + tile];
  if (scene < 0) return;                         // uniform exit, EXEC stays full
  constexpr int NTILE = 64 << TPW;
  const int nbase = blockIdx.y * NTILE;
  const int tid   = threadIdx.x;
  const int lane  = tid & 31;
  const int wave  = tid >> 5;
  const int m     = lane & 15;
  const int khalf = lane >> 4;
  const int* rowmap = meta + M_ROWMAP;

  // A-tile load coordinates: 2 x b128 per thread (16 rows x 64 cols)
  int aR[2], aC[2];
  const float* aG[2];
  #pragma unroll
  for (int i = 0; i < 2; ++i) {
    int idx = tid + i * 128;
    aR[i] = idx >> 4;
    aC[i] = (idx & 15) * 4;
    int prow = tile * 16 + aR[i];
    int arow = GATHER ? rowmap[prow] : prow;
    if (arow < 0) arow = 0;                      // pad row: garbage, never stored
    aG[i] = A + (size_t)arow * Kdim + aC[i];
  }
  // B-tile: (8<<TPW) x b128 per thread (64 rows x NTILE cols)
  constexpr int B_ITERS  = 8 << TPW;
  constexpr int C4_MASK  = (16 << TPW) - 1;
  constexpr int KR_SHIFT = 4 + TPW;
  const float* WsBase = Ws + (size_t)scene * Kdim * Ndim + nbase;

  const int numChunks = Kdim >> 6;

  auto prefetch = [&](int buf, int kc) {
    #pragma unroll
    for (int i = 0; i < 2; ++i)
      async_b128(aG[i] + kc, &sA[buf][aR[i]][aC[i]]);
    #pragma unroll
    for (int i = 0; i < B_ITERS; ++i) {
      int idx = tid + i * 128;
      int kr  = idx >> KR_SHIFT;
      int c4  = (idx & C4_MASK) * 4;
      async_b128(WsBase + (size_t)(kc + kr) * Ndim + c4, &sB[buf][kr][c4]);
    }
  };

  v8f acc0 = {}, acc1 = {};

  prefetch(0, 0);
  for (int c = 0; c < numChunks; ++c) {
    wait_async0();                               // this wave's chunk-c loads done
    __syncthreads();                             // all loads visible; buf c+1 free
    if (c + 1 < numChunks) prefetch((c + 1) & 1, (c + 1) << 6);
    const int buf = c & 1;
    #pragma unroll
    for (int k4 = 0; k4 < 16; ++k4) {
      const int kb = k4 * 4 + khalf * 2;
      v2f av; av.x = sA[buf][m][kb]; av.y = sA[buf][m][kb + 1];
      v2f b0; b0.x = sB[buf][kb][wave * 16 + m];
      b0.y = sB[buf][kb + 1][wave * 16 + m];
      acc0 = __builtin_amdgcn_wmma_f32_16x16x4_f32(
          false, av, false, b0, (short)0, acc0, false, false);
      if (TPW) {
        v2f b1; b1.x = sB[buf][kb][64 + wave * 16 + m];
        b1.y = sB[buf][kb + 1][64 + wave * 16 + m];
        acc1 = __builtin_amdgcn_wmma_f32_16x16x4_f32(
            false, av, false, b1, (short)0, acc1, false, false);
      }
    }
    __syncthreads();                             // done reading buf before refill
  }

  // epilogue: bias + global bias, optional relu, store (scatter for last layer)
  auto store_tile = [&](v8f acc, int col) {
    float bias = b[(size_t)scene * Ndim + col] + gb[col];
    #pragma unroll
    for (int r = 0; r < 8; ++r) {
      float v = acc[r] + bias;
      if (RELU) v = fmaxf(v, 0.0f);
      int mrow = r + khalf * 8;                  // C/D layout: vgpr r -> M=r(+8)
      int prow = tile * 16 + mrow;
      if (SCATTER) {
        int g = rowmap[prow];
        if (g >= 0) Out[(size_t)g * Ndim + col] = v;
      } else {
        Out[(size_t)prow * Ndim + col] = v;
      }
    }
  };
  const int col0 = nbase + wave * 16 + m;
  store_tile(acc0, col0);
  if (TPW) store_tile(acc1, col0 + 64);
}

// ---------------- launcher ----------------
extern "C" void kernel_launch(void* const* d_in, const int* in_sizes, int n_in,
                              void* d_out, int out_size, void* d_ws, size_t ws_size,
                              hipStream_t stream) {
  const float* x     = (const float*)d_in[0];
  const int*   scene = (const int*)d_in[1];
  const float* W0    = (const float*)d_in[2];
  const float* b0    = (const float*)d_in[3];
  const float* gW0   = (const float*)d_in[4];
  const float* gb0   = (const float*)d_in[5];
  const float* W1    = (const float*)d_in[6];
  const float* b1    = (const float*)d_in[7];
  const float* gW1   = (const float*)d_in[8];
  const float* gb1   = (const float*)d_in[9];
  const float* W2    = (const float*)d_in[10];
  const float* b2    = (const float*)d_in[11];
  const float* gW2   = (const float*)d_in[12];
  const float* gb2   = (const float*)d_in[13];

  int*   meta = (int*)d_ws;
  float* H1   = (float*)((char*)d_ws + H1_OFF);
  float* H2   = (float*)((char*)d_ws + H2_OFF);
  float* Ws0  = (float*)((char*)d_ws + WS0_OFF);
  float* Ws1  = (float*)((char*)d_ws + WS1_OFF);
  float* Ws2  = (float*)((char*)d_ws + WS2_OFF);
  float* out  = (float*)d_out;

  k_init   <<<(MAXP + 255) / 256, 256, 0, stream>>>(meta);
  k_hist   <<<BATCH / 256, 256, 0, stream>>>(meta, scene);
  k_plan   <<<1, 32, 0, stream>>>(meta);
  k_scatter<<<BATCH / 256, 256, 0, stream>>>(meta, scene);

  const int t0 = NS * 512 * 512, t1 = NS * 512 * 256, t2 = NS * 256 * 64;
  k_scale<<<(t0 + 255) / 256, 256, 0, stream>>>(W0, gW0, Ws0, 512 * 512, t0);
  k_scale<<<(t1 + 255) / 256, 256, 0, stream>>>(W1, gW1, Ws1, 512 * 256, t1);
  k_scale<<<(t2 + 255) / 256, 256, 0, stream>>>(W2, gW2, Ws2, 256 * 64, t2);

  // layer 0: 512->512, gather rows, relu, 128-col blocks
  k_star_gemm<1, 1, 0, 1><<<dim3(MAX_TILES, 4), 128, 0, stream>>>(
      x, Ws0, b0, gb0, H1, meta, 512, 512);
  // layer 1: 512->256, permuted in/out, relu, 128-col blocks
  k_star_gemm<1, 0, 0, 1><<<dim3(MAX_TILES, 2), 128, 0, stream>>>(
      H1, Ws1, b1, gb1, H2, meta, 512, 256);
  // layer 2: 256->64, scatter back, no relu, 64-col blocks
  k_star_gemm<0, 0, 1, 0><<<dim3(MAX_TILES, 1), 128, 0, stream>>>(
      H2, Ws2, b2, gb2, out, meta, 256, 64);
}